// EntityLinker_60490319397253
// MI455X (gfx1250) — compile-verified
//
#include <hip/hip_runtime.h>
#include <math.h>

typedef __attribute__((ext_vector_type(2))) float v2f;
typedef __attribute__((ext_vector_type(8))) float v8f;

#define DD   128
#define QQ   64
#define CC   64
#define TT   8
#define S128 132   // padded LDS stride for 128-wide rows (132 % 64 == 4 -> conflict-free column reads)
#define S64  68    // padded LDS stride for 64-wide rows

__device__ __forceinline__ v8f wmma4(v2f a, v2f b, v8f c) {
  // D(16x16,f32) += A(16x4,f32) * B(4x16,f32), fp32-exact
  return __builtin_amdgcn_wmma_f32_16x16x4_f32(
      /*neg_a=*/false, a, /*neg_b=*/false, b,
      /*c_mod=*/(short)0, c, /*reuse_a=*/false, /*reuse_b=*/false);
}

__global__ __launch_bounds__(256)
void entity_linker_kernel(const int* __restrict__ q_ids,
                          const int* __restrict__ c_ids,
                          const int* __restrict__ num_qs,
                          const float* __restrict__ embed,
                          const float* __restrict__ W_h,
                          const float* __restrict__ b_h,
                          const float* __restrict__ W_o,
                          const float* __restrict__ b_o,
                          float* __restrict__ out) {
  extern __shared__ float smem[];
  float* sq   = smem;               // [64][S128] q_h (reused later for h)
  float* sc   = sq  + 64 * S128;    // [64][S128] c_h
  float* swq  = sc  + 64 * S128;    // [64][S128] weighted_q
  float* satt = swq + 64 * S128;    // [64][S64]  sim / att
  float* qsum = satt + 64 * S64;    // [128]      q_summary

  const int b    = blockIdx.x;
  const int tid  = threadIdx.x;
  const int wave = tid >> 5;
  const int lane = tid & 31;
  const int half = lane >> 4;
  const int l    = lane & 15;
  const int num_q = num_qs[b];

  // ---- gather q_h: thread -> (q row, 32-float slab) ----
  {
    const int q  = tid >> 2;
    const int d0 = (tid & 3) * 32;
    const float* src = embed + (size_t)q_ids[(size_t)b * QQ + q] * DD + d0;
    float* dst = sq + q * S128 + d0;
#pragma unroll
    for (int i = 0; i < 32; ++i) dst[i] = src[i];
  }
  // ---- gather + pad-masked mean c_h ----
  {
    const int c  = tid >> 2;
    const int d0 = (tid & 3) * 32;
    const int* ids = c_ids + ((size_t)b * CC + c) * TT;
    float acc[32];
#pragma unroll
    for (int i = 0; i < 32; ++i) acc[i] = 0.f;
    float cnt = 0.f;
    for (int t = 0; t < TT; ++t) {
      const int id = ids[t];
      if (id != 0) {
        cnt += 1.f;
        const float* e = embed + (size_t)id * DD + d0;
#pragma unroll
        for (int i = 0; i < 32; ++i) acc[i] += e[i];
      }
    }
    const float inv = 1.f / fmaxf(cnt, 1.f);
    float* dst = sc + c * S128 + d0;
#pragma unroll
    for (int i = 0; i < 32; ++i) dst[i] = acc[i] * inv;
  }
  __syncthreads();

  // ---- q_summary: length-masked mean over Q ----
  if (tid < DD) {
    float s = 0.f;
    for (int q = 0; q < num_q; ++q) s += sq[q * S128 + tid];
    qsum[tid] = s / fmaxf((float)num_q, 1.f);
  }
  __syncthreads();

  // ---- sim = c_h @ q_h^T / sqrt(D), 16 tiles, 2 per wave ----
  const float scale = 0.088388347648318447f; // 1/sqrt(128)
#pragma unroll
  for (int t2 = 0; t2 < 2; ++t2) {
    const int t  = wave * 2 + t2;
    const int mt = t >> 2, nt = t & 3;
    const float* Ar = sc + (mt * 16 + l) * S128;  // A row of c_h
    const float* Br = sq + (nt * 16 + l) * S128;  // B = q_h^T -> lane reads q_h row
    v8f acc = {};
    for (int kk = 0; kk < 32; ++kk) {
      const int k0 = kk * 4 + 2 * half;
      v2f a;  a.x  = Ar[k0]; a.y  = Ar[k0 + 1];
      v2f bv; bv.x = Br[k0]; bv.y = Br[k0 + 1];
      acc = wmma4(a, bv, acc);
    }
    const int ncol = nt * 16 + l;
    const bool valid = (ncol < num_q);
#pragma unroll
    for (int r = 0; r < 8; ++r) {
      const int row = mt * 16 + r + 8 * half;
      satt[row * S64 + ncol] = valid ? acc[r] * scale : -__builtin_inff();
    }
  }
  __syncthreads();

  // ---- masked softmax over q, one row per thread ----
  if (tid < CC) {
    float* row = satt + tid * S64;
    float m = -__builtin_inff();
    for (int q = 0; q < QQ; ++q) m = fmaxf(m, row[q]);
    float s = 0.f;
    for (int q = 0; q < QQ; ++q) { const float e = __expf(row[q] - m); row[q] = e; s += e; }
    const float inv = 1.f / s;
    for (int q = 0; q < QQ; ++q) row[q] *= inv;
  }
  __syncthreads();

  // ---- weighted_q = att @ q_h, 32 tiles, 4 per wave ----
#pragma unroll
  for (int t4 = 0; t4 < 4; ++t4) {
    const int t  = wave * 4 + t4;
    const int mt = t >> 3, nt = t & 7;
    const float* Ar = satt + (mt * 16 + l) * S64;
    const int ncol = nt * 16 + l;
    v8f acc = {};
    for (int kk = 0; kk < 16; ++kk) {
      const int k0 = kk * 4 + 2 * half;
      v2f a;  a.x  = Ar[k0];                 a.y  = Ar[k0 + 1];
      v2f bv; bv.x = sq[k0 * S128 + ncol];   bv.y = sq[(k0 + 1) * S128 + ncol];
      acc = wmma4(a, bv, acc);
    }
#pragma unroll
    for (int r = 0; r < 8; ++r)
      swq[(mt * 16 + r + 8 * half) * S128 + ncol] = acc[r];
  }
  __syncthreads();

  // ---- h = tanh(feats @ W_h + b_h): K=640 split into 5 branch-free 128-wide segments ----
#pragma unroll
  for (int t4 = 0; t4 < 4; ++t4) {
    const int t  = wave * 4 + t4;
    const int mt = t >> 3, nt = t & 7;
    const int arow = mt * 16 + l;
    const int ncol = nt * 16 + l;
    const float* Cr = sc  + arow * S128;   // c_h row
    const float* Qr = swq + arow * S128;   // weighted_q row
    const float* Wp = W_h + ncol;          // W_h[k][ncol] == Wp[k * DD]
    v8f acc = {};
    // seg 0: q_summary tiled (k = 0..127)
    for (int kk = 0; kk < 32; ++kk) {
      const int k0 = kk * 4 + 2 * half;
      v2f a;  a.x  = qsum[k0];                a.y  = qsum[k0 + 1];
      v2f bv; bv.x = Wp[(size_t)k0 * DD];     bv.y = Wp[(size_t)(k0 + 1) * DD];
      acc = wmma4(a, bv, acc);
    }
    // seg 1: c_h (k = 128..255)
    for (int kk = 0; kk < 32; ++kk) {
      const int k0 = kk * 4 + 2 * half;
      v2f a;  a.x  = Cr[k0];                       a.y  = Cr[k0 + 1];
      v2f bv; bv.x = Wp[(size_t)(k0 + 128) * DD];  bv.y = Wp[(size_t)(k0 + 129) * DD];
      acc = wmma4(a, bv, acc);
    }
    // seg 2: weighted_q (k = 256..383)
    for (int kk = 0; kk < 32; ++kk) {
      const int k0 = kk * 4 + 2 * half;
      v2f a;  a.x  = Qr[k0];                       a.y  = Qr[k0 + 1];
      v2f bv; bv.x = Wp[(size_t)(k0 + 256) * DD];  bv.y = Wp[(size_t)(k0 + 257) * DD];
      acc = wmma4(a, bv, acc);
    }
    // seg 3: c_h * weighted_q (k = 384..511)
    for (int kk = 0; kk < 32; ++kk) {
      const int k0 = kk * 4 + 2 * half;
      v2f a;  a.x  = Cr[k0] * Qr[k0];              a.y  = Cr[k0 + 1] * Qr[k0 + 1];
      v2f bv; bv.x = Wp[(size_t)(k0 + 384) * DD];  bv.y = Wp[(size_t)(k0 + 385) * DD];
      acc = wmma4(a, bv, acc);
    }
    // seg 4: |c_h - weighted_q| (k = 512..639)
    for (int kk = 0; kk < 32; ++kk) {
      const int k0 = kk * 4 + 2 * half;
      v2f a;  a.x  = fabsf(Cr[k0] - Qr[k0]);       a.y  = fabsf(Cr[k0 + 1] - Qr[k0 + 1]);
      v2f bv; bv.x = Wp[(size_t)(k0 + 512) * DD];  bv.y = Wp[(size_t)(k0 + 513) * DD];
      acc = wmma4(a, bv, acc);
    }
    const float bias = b_h[ncol];
#pragma unroll
    for (int r = 0; r < 8; ++r)
      sq[(mt * 16 + r + 8 * half) * S128 + ncol] = tanhf(acc[r] + bias);  // reuse q_h LDS for h
  }
  __syncthreads();

  // ---- out = h @ W_o + b_o ----
  if (tid < CC) {
    float s = 0.f;
    for (int n = 0; n < DD; ++n) s += sq[tid * S128 + n] * W_o[n];
    out[(size_t)b * CC + tid] = s + b_o[0];
  }
}

extern "C" void kernel_launch(void* const* d_in, const int* in_sizes, int n_in,
                              void* d_out, int out_size, void* d_ws, size_t ws_size,
                              hipStream_t stream) {
  (void)in_sizes; (void)n_in; (void)out_size; (void)d_ws; (void)ws_size;
  const int*   q_ids  = (const int*)d_in[0];
  const int*   c_ids  = (const int*)d_in[1];
  const int*   num_qs = (const int*)d_in[2];
  // d_in[3] = num_cols (unused by the reference output)
  const float* embed  = (const float*)d_in[4];
  const float* W_h    = (const float*)d_in[5];
  const float* b_h    = (const float*)d_in[6];
  const float* W_o    = (const float*)d_in[7];
  const float* b_o    = (const float*)d_in[8];
  float* out = (float*)d_out;

  const size_t shmem = (size_t)(3 * 64 * S128 + 64 * S64 + 128) * sizeof(float); // ~117 KB
  hipFuncSetAttribute(reinterpret_cast<const void*>(entity_linker_kernel),
                      hipFuncAttributeMaxDynamicSharedMemorySize, (int)shmem);
  entity_linker_kernel<<<1024, 256, shmem, stream>>>(
      q_ids, c_ids, num_qs, embed, W_h, b_h, W_o, b_o, out);
}